// SceneGuidance_61787399520371
// MI455X (gfx1250) — compile-verified
//
#include <hip/hip_runtime.h>

#define GRID_N 2048
#define NPIX (GRID_N * GRID_N)

typedef __attribute__((ext_vector_type(2))) float v2f;
typedef __attribute__((ext_vector_type(8))) float v8f;

// Matches JAX _world_to_grid: clip, subtract xmin, divide by f32 DX, trunc.
__device__ __forceinline__ int world_to_grid_1d(float c) {
    const float DXf = (float)(200.0 / 2047.0);
    float cc = fminf(fmaxf(c, -100.0f), 100.0f);
    int g = (int)((cc + 100.0f) / DXf);
    g = g < 0 ? 0 : g;
    g = g > (GRID_N - 1) ? (GRID_N - 1) : g;
    return g;
}

__global__ __launch_bounds__(256) void k_init(float4* __restrict__ b) {
    int i = blockIdx.x * 256 + threadIdx.x;
    if (i < NPIX / 4) b[i] = make_float4(0.5f, 0.5f, 0.5f, 0.5f);
}

__global__ __launch_bounds__(256) void k_scatter_map(const float* __restrict__ xy,
                                                     const int* __restrict__ valid,
                                                     const int* __restrict__ types,
                                                     float* __restrict__ b, int n) {
    int i = blockIdx.x * 256 + threadIdx.x;
    if (i >= n) return;
    // min(0.5, 1.0) is a no-op; only solid (valid && LANE_BOUNDARY) points matter.
    if (valid[i] != 0 && types[i] == 1) {
        int gx = world_to_grid_1d(xy[2 * i + 0]);
        int gy = world_to_grid_1d(xy[2 * i + 1]);
        b[gy * GRID_N + gx] = 0.0f;   // min(0.5, 0.0): race-benign, all write 0
    }
}

__global__ __launch_bounds__(256) void k_scatter_obs(const float* __restrict__ xy,
                                                     float* __restrict__ b, int n) {
    int i = blockIdx.x * 256 + threadIdx.x;
    if (i >= n) return;
    int gx = world_to_grid_1d(xy[2 * i + 0]);
    int gy = world_to_grid_1d(xy[2 * i + 1]);
    b[gy * GRID_N + gx] = 0.0f;
}

__global__ void k_goal(const float* __restrict__ goal, float* __restrict__ b) {
    int gx = world_to_grid_1d(goal[0]);
    int gy = world_to_grid_1d(goal[1]);
    b[gy * GRID_N + gx] = 1.0f;       // .set() runs after the mins (stream order)
}

// One Jacobi sweep. One wave32 per 16x16 tile.
// lap = S*U + U*S  (S = tridiagonal shift matrix) via 8x v_wmma_f32_16x16x4_f32,
// halo rows/columns folded into the C accumulator before the WMMA chain.
__global__ __launch_bounds__(256) void k_jacobi(const float* __restrict__ u,
                                                const float* __restrict__ bgrid,
                                                float* __restrict__ out) {
    const int lane = threadIdx.x & 31;
    const int wave = threadIdx.x >> 5;
    const int w    = blockIdx.x * 8 + wave;      // 2048 blocks * 8 waves = 16384 tiles
    const int tx   = w & 127;
    const int ty   = w >> 7;
    const int x0   = tx << 4;
    const int y0   = ty << 4;
    const int n    = lane & 15;                  // column within tile (B/C layout)
    const int hi   = lane >> 4;                  // upper half of wave

    // ---- operand construction ----------------------------------------------
    // A f32 16x4 layout: VGPR v holds K = v + 2*hi  (ISA 7.12.2); B 4x16 mirrors it.
    v2f ub[4];    // U rows 4k..4k+3 in B layout  (for S*U)
    v2f ua[4];    // U cols 4k..4k+3 in A layout  (for U*S)
    v2f sm[4];    // S chunks (same expression works as A-chunk and B-chunk: S symmetric)
#pragma unroll
    for (int k = 0; k < 4; ++k) {
#pragma unroll
        for (int v = 0; v < 2; ++v) {
            int Kloc = v + 2 * hi;
            int col  = 4 * k + Kloc;
            ub[k][v] = u[(y0 + col) * GRID_N + x0 + n];
            ua[k][v] = u[(y0 + n) * GRID_N + x0 + col];
            int d = n - col;
            sm[k][v] = (d * d == 1) ? 1.0f : 0.0f;
        }
    }

    // ---- halo values --------------------------------------------------------
    // lanes 0-15: left halo column rows 0-15; lanes 16-31: right halo column rows 0-15
    int   ccol   = hi ? (x0 + 16) : (x0 - 1);
    bool  cok    = hi ? (tx < 127) : (tx > 0);
    float colval = cok ? u[(y0 + n) * GRID_N + ccol] : 0.0f;
    float upv = (hi == 0 && ty > 0)   ? u[(y0 - 1) * GRID_N + x0 + n]  : 0.0f;
    float dnv = (hi == 1 && ty < 127) ? u[(y0 + 16) * GRID_N + x0 + n] : 0.0f;

    // ---- C accumulator = halo contributions (C layout: row = c + 8*hi) ------
    v8f acc;
#pragma unroll
    for (int c = 0; c < 8; ++c) {
        int   rl = c + 8 * hi;
        float lv = __shfl(colval, rl, 32);        // left halo for this row
        float rv = __shfl(colval, 16 + rl, 32);   // right halo for this row
        float init = 0.0f;
        if (c == 0) init += upv;                  // upv == 0 in hi half
        if (c == 7) init += dnv;                  // dnv == 0 in lo half
        init += (n == 0)  ? lv : 0.0f;
        init += (n == 15) ? rv : 0.0f;
        acc[c] = init;
    }

    // ---- 8 chained WMMAs: acc += S*U + U*S ---------------------------------
#pragma unroll
    for (int k = 0; k < 4; ++k)
        acc = __builtin_amdgcn_wmma_f32_16x16x4_f32(false, sm[k], false, ub[k],
                                                    (short)0, acc, false, false);
#pragma unroll
    for (int k = 0; k < 4; ++k)
        acc = __builtin_amdgcn_wmma_f32_16x16x4_f32(false, ua[k], false, sm[k],
                                                    (short)0, acc, false, false);

    // ---- select boundary vs relaxed value, store (C layout, coalesced halves)
    // Multiply by compile-time reciprocal: avoids the 15-op IEEE divide sequence
    // + exec-branch per element that a literal division generates.
    const float inv_denom =
        (float)(1.0 / (4.0 + (200.0 / 2047.0) * (200.0 / 2047.0)));
#pragma unroll
    for (int c = 0; c < 8; ++c) {
        int   idx = (y0 + c + 8 * hi) * GRID_N + x0 + n;
        float bv  = bgrid[idx];                   // boundary cells hold b forever
        out[idx]  = (bv != 0.5f) ? bv : (acc[c] * inv_denom);
    }
}

// Bilinear sample 5120 trajectory points, block-reduce, write -sum.
__global__ __launch_bounds__(256) void k_sample(const float* __restrict__ traj,
                                                const float* __restrict__ field,
                                                float* __restrict__ out, int npts) {
    __shared__ float red[256];
    float s = 0.0f;
    for (int i = threadIdx.x; i < npts; i += 256) {
        float x = traj[2 * i + 0];
        float y = traj[2 * i + 1];
        float xn = (x + 100.0f) / 200.0f * 2.0f - 1.0f;
        float yn = (y + 100.0f) / 200.0f * 2.0f - 1.0f;
        float px = fminf(fmaxf((xn + 1.0f) * 0.5f * 2047.0f, 0.0f), 2047.0f);
        float py = fminf(fmaxf((yn + 1.0f) * 0.5f * 2047.0f, 0.0f), 2047.0f);
        int ix0 = (int)floorf(px), iy0 = (int)floorf(py);
        int ix1 = min(ix0 + 1, 2047), iy1 = min(iy0 + 1, 2047);
        float wx = px - (float)ix0, wy = py - (float)iy0;
        float v00 = field[iy0 * GRID_N + ix0], v01 = field[iy0 * GRID_N + ix1];
        float v10 = field[iy1 * GRID_N + ix0], v11 = field[iy1 * GRID_N + ix1];
        s += (v00 * (1.0f - wx) + v01 * wx) * (1.0f - wy)
           + (v10 * (1.0f - wx) + v11 * wx) * wy;
    }
    red[threadIdx.x] = s;
    __syncthreads();
    for (int off = 128; off > 0; off >>= 1) {
        if (threadIdx.x < off) red[threadIdx.x] += red[threadIdx.x + off];
        __syncthreads();
    }
    if (threadIdx.x == 0) out[0] = -red[0];
}

extern "C" void kernel_launch(void* const* d_in, const int* in_sizes, int n_in,
                              void* d_out, int out_size, void* d_ws, size_t ws_size,
                              hipStream_t stream) {
    const float* traj      = (const float*)d_in[0];   // (64,80,2)
    const float* goal      = (const float*)d_in[1];   // (2,)
    const float* obstacles = (const float*)d_in[2];   // (256,2)
    const float* map_xy    = (const float*)d_in[3];   // (100000,2)
    const int*   map_valid = (const int*)d_in[4];     // (100000,)
    const int*   map_types = (const int*)d_in[5];     // (100000,)
    float*       out       = (float*)d_out;

    float* b  = (float*)d_ws;                         // boundary grid (persistent)
    float* u0 = b + NPIX;                             // ping
    float* u1 = u0 + NPIX;                            // pong

    int n_map = in_sizes[4];
    int n_obs = in_sizes[2] / 2;
    int n_pts = in_sizes[0] / 2;

    k_init<<<(NPIX / 4 + 255) / 256, 256, 0, stream>>>((float4*)b);
    k_scatter_map<<<(n_map + 255) / 256, 256, 0, stream>>>(map_xy, map_valid, map_types, b, n_map);
    k_scatter_obs<<<(n_obs + 255) / 256, 256, 0, stream>>>(obstacles, b, n_obs);
    k_goal<<<1, 1, 0, stream>>>(goal, b);

    const float* src = b;                             // iteration 0 reads b directly
    float*       dst = u0;
    for (int it = 0; it < 64; ++it) {
        k_jacobi<<<2048, 256, 0, stream>>>(src, b, dst);
        src = dst;
        dst = (dst == u0) ? u1 : u0;
    }
    k_sample<<<1, 256, 0, stream>>>(traj, src, out, n_pts);
}